// droneDeliveryModel_31327491457450
// MI455X (gfx1250) — compile-verified
//
#include <hip/hip_runtime.h>
#include <hip/hip_bf16.h>

typedef float v2f __attribute__((ext_vector_type(2)));
typedef float v8f __attribute__((ext_vector_type(8)));

#define N_NODES 100000
#define N_EDGES 1600000
#define C_HID   32
#define C_OUT   8
#define N_AGENTS 1000
#define N_TILES (N_NODES / 16)   // 6250, exact

// ---------------------------------------------------------------- zero fill
__global__ void zero_f32(float* __restrict__ p, int n) {
    int t = blockIdx.x * blockDim.x + threadIdx.x;
    if (t < n) p[t] = 0.0f;
}

// ---------------------------------------------------------------- degree count
__global__ void count_deg(const int* __restrict__ dst, float* __restrict__ deg) {
    int t = blockIdx.x * blockDim.x + threadIdx.x;
    if (t < N_EDGES) atomicAdd(deg + dst[t], 1.0f);
}

// ---------------------------------------------------------------- edge scatter-add
// one thread per (edge, 4-channel group): float4 gather + 4 f32 atomics
__global__ __launch_bounds__(256) void scatter_add(
        const float* __restrict__ xin, const int* __restrict__ src,
        const int* __restrict__ dst, float* __restrict__ agg) {
    int t = blockIdx.x * blockDim.x + threadIdx.x;
    if (t >= N_EDGES * 8) return;
    int e = t >> 3;
    int c = (t & 7) * 4;
    int s = src[e];
    int d = dst[e];
    const float4 v = *(const float4*)(xin + (size_t)s * C_HID + c);
    float* p = agg + (size_t)d * C_HID + c;
    atomicAdd(p + 0, v.x);
    atomicAdd(p + 1, v.y);
    atomicAdd(p + 2, v.z);
    atomicAdd(p + 3, v.w);
}

// ---------------------------------------------------------------- fused SAGE GEMM
// hout[16x32 tile] = relu( (agg/max(deg,1)) @ Wl + xin @ Wr + bias )
// One wave per 16-node tile. K=64 (agg||xin), via 16 chunks of K=4 fp32 WMMA,
// two 16x16 N-tiles -> 32 v_wmma_f32_16x16x4_f32 per wave.
__global__ __launch_bounds__(256) void sage_gemm(
        const float* __restrict__ agg, const float* __restrict__ xin,
        const float* __restrict__ deg,
        const float* __restrict__ Wl, const float* __restrict__ Wr,
        const float* __restrict__ bias, float* __restrict__ hout) {
    __shared__ float sW[64 * 32];   // [k][n], k<32 = Wl, k>=32 = Wr
    __shared__ float sB[32];
    for (int i = threadIdx.x; i < 1024; i += 256) {
        sW[i]        = Wl[i];
        sW[1024 + i] = Wr[i];
    }
    if (threadIdx.x < 32) sB[threadIdx.x] = bias[threadIdx.x];
    __syncthreads();

    const int wave = threadIdx.x >> 5;
    const int lane = threadIdx.x & 31;
    const int tile = blockIdx.x * 8 + wave;
    if (tile >= N_TILES) return;          // wave-uniform: EXEC stays all-1s

    const int row0 = tile * 16;
    const int mrow = lane & 15;           // A/B row/col index within tile
    const int row  = row0 + mrow;
    const int koff = (lane >> 4) * 2;     // lane-half K offset (A & B layout)

    const float dv  = deg[row];
    const float inv = 1.0f / fmaxf(dv, 1.0f);

    v8f acc0, acc1;
    const float b0v = sB[mrow];
    const float b1v = sB[16 + mrow];
#pragma unroll
    for (int j = 0; j < 8; ++j) { acc0[j] = b0v; acc1[j] = b1v; }

    const float* aggRow = agg + (size_t)row * C_HID;
    const float* xRow   = xin + (size_t)row * C_HID;

#pragma unroll
    for (int kc = 0; kc < 16; ++kc) {
        const int kbase = kc * 4;
        const int kb = kbase + koff;      // concatenated-K index 0..63
        v2f a;
        if (kc < 8) {                     // mean-aggregated neighbor features
            a.x = aggRow[kb]     * inv;
            a.y = aggRow[kb + 1] * inv;
        } else {                          // root/self features
            a.x = xRow[kb - 32];
            a.y = xRow[kb - 31];
        }
        v2f b0, b1;
        b0.x = sW[(kb)     * 32 + mrow];
        b0.y = sW[(kb + 1) * 32 + mrow];
        b1.x = sW[(kb)     * 32 + 16 + mrow];
        b1.y = sW[(kb + 1) * 32 + 16 + mrow];
        acc0 = __builtin_amdgcn_wmma_f32_16x16x4_f32(
                   false, a, false, b0, (short)0, acc0, false, false);
        acc1 = __builtin_amdgcn_wmma_f32_16x16x4_f32(
                   false, a, false, b1, (short)0, acc1, false, false);
    }

    // C/D layout: VGPR j -> M = j + (lane>>4)*8, N = lane&15
    const int half = lane >> 4;
#pragma unroll
    for (int j = 0; j < 8; ++j) {
        const int m = row0 + half * 8 + j;
        hout[(size_t)m * C_HID + mrow]      = fmaxf(acc0[j], 0.0f);
        hout[(size_t)m * C_HID + 16 + mrow] = fmaxf(acc1[j], 0.0f);
    }
}

// ---------------------------------------------------------------- output head
__global__ void out_proj(const float* __restrict__ h, const float* __restrict__ Wo,
                         const float* __restrict__ bo, float* __restrict__ out) {
    int t = blockIdx.x * blockDim.x + threadIdx.x;
    if (t >= N_AGENTS * C_OUT) return;
    const int n = t >> 3, o = t & 7;
    float s = bo[o];
    const float* hr = h + (size_t)n * C_HID;
#pragma unroll
    for (int k = 0; k < C_HID; ++k) s += hr[k] * Wo[k * C_OUT + o];
    out[t] = s;
}

// ---------------------------------------------------------------- launcher
extern "C" void kernel_launch(void* const* d_in, const int* in_sizes, int n_in,
                              void* d_out, int out_size, void* d_ws, size_t ws_size,
                              hipStream_t stream) {
    const float* x    = (const float*)d_in[0];
    const int*   ei   = (const int*)d_in[1];       // [2, E]
    const int*   src  = ei;
    const int*   dst  = ei + N_EDGES;
    const float* Wl1  = (const float*)d_in[2];
    const float* bl1  = (const float*)d_in[3];
    const float* Wr1  = (const float*)d_in[4];
    const float* Wl2  = (const float*)d_in[5];
    const float* bl2  = (const float*)d_in[6];
    const float* Wr2  = (const float*)d_in[7];
    const float* Wl3  = (const float*)d_in[8];
    const float* bl3  = (const float*)d_in[9];
    const float* Wr3  = (const float*)d_in[10];
    const float* Wo   = (const float*)d_in[11];
    const float* bo   = (const float*)d_in[12];
    float* out = (float*)d_out;

    const size_t NB = (size_t)N_NODES * C_HID;     // elements per feature buffer
    float* agg = (float*)d_ws;
    float* h1  = agg + NB;
    float* h2  = h1 + NB;
    float* deg = h2 + NB;

    const int ZB_FEAT = (int)((NB + 255) / 256);
    const int ZB_DEG  = (N_NODES + 255) / 256;
    const int EB      = (N_EDGES + 255) / 256;
    const int SB      = (N_EDGES * 8 + 255) / 256;
    const int GB      = (N_TILES + 7) / 8;         // 8 waves/block, 1 tile/wave
    const int OB      = (N_AGENTS * C_OUT + 255) / 256;

    // degrees (once)
    zero_f32<<<ZB_DEG, 256, 0, stream>>>(deg, N_NODES);
    count_deg<<<EB, 256, 0, stream>>>(dst, deg);

    // layer 1: x -> h1
    zero_f32<<<ZB_FEAT, 256, 0, stream>>>(agg, (int)NB);
    scatter_add<<<SB, 256, 0, stream>>>(x, src, dst, agg);
    sage_gemm<<<GB, 256, 0, stream>>>(agg, x, deg, Wl1, Wr1, bl1, h1);

    // layer 2: h1 -> h2
    zero_f32<<<ZB_FEAT, 256, 0, stream>>>(agg, (int)NB);
    scatter_add<<<SB, 256, 0, stream>>>(h1, src, dst, agg);
    sage_gemm<<<GB, 256, 0, stream>>>(agg, h1, deg, Wl2, Wr2, bl2, h2);

    // layer 3: h2 -> h1 (reuse)
    zero_f32<<<ZB_FEAT, 256, 0, stream>>>(agg, (int)NB);
    scatter_add<<<SB, 256, 0, stream>>>(h2, src, dst, agg);
    sage_gemm<<<GB, 256, 0, stream>>>(agg, h2, deg, Wl3, Wr3, bl3, h1);

    // output head: first 1000 nodes
    out_proj<<<OB, 256, 0, stream>>>(h1, Wo, bo, out);
}